// BoundaryAwareDummyCausalLM_17798344474794
// MI455X (gfx1250) — compile-verified
//
#include <hip/hip_runtime.h>
#include <hip/hip_bf16.h>
#include <stdint.h>

#define VOCAB 32000
#define NEGVAL -1000.0f

// CDNA5 async LDS->global store (ASYNCcnt-tracked store engine).
// gfx1250 VGLOBAL async encoding: vaddr (64-bit VGPR pair), LDS-addr VGPR, off.
__device__ __forceinline__ void async_store_b128_nt(unsigned long long gaddr,
                                                    unsigned lds_off) {
  asm volatile("global_store_async_from_lds_b128 %0, %1, off th:TH_STORE_NT"
               :: "v"(gaddr), "v"(lds_off)
               : "memory");
}

__device__ __forceinline__ void wait_async_0() {
  asm volatile("s_wait_asynccnt 0" ::: "memory");
}

__global__ __launch_bounds__(256) void BoundaryAwareDummyCausalLM_row_fill(
    const long long* __restrict__ input_ids,      // (B,T) int64
    const int*       __restrict__ attention_mask, // (B,T) int32
    const long long* __restrict__ start_positions,// (B,)  int64
    float*           __restrict__ out)            // (B,T,VOCAB) f32
{
  __shared__ float4 tile[256];   // 4 KB: one uniform-fill float4 slot per thread

  const int p = blockIdx.x;          // position in sequence, 0..T-1
  const int b = blockIdx.y;          // batch index
  const int T = gridDim.x;
  const int row = b * T + p;

  // Per-row constant fill value + target column to force to zero.
  float fill = 0.0f;
  int   tgt  = -1;
  if (p != T - 1) {
    const long long sp = start_positions[b];
    const bool m = (attention_mask[b * T + p + 1] != 0) &&
                   ((long long)(p + 1) >= sp);
    fill = m ? NEGVAL : 0.0f;
    tgt  = (int)input_ids[b * T + p + 1];   // 0..VOCAB-1
  }

  const float4 fv = make_float4(fill, fill, fill, fill);
  tile[threadIdx.x] = fv;
  __syncthreads();   // DS writes visible before the async engine reads LDS

  // 32-bit wave-relative LDS byte offset (flat shared addr truncates to it).
  const unsigned lds_off = (unsigned)(uintptr_t)(&tile[threadIdx.x]);

  float* rowp = out + (size_t)row * VOCAB;
  const int F4 = VOCAB / 4;                         // 8000 float4s per row

  // Pure streaming loop: one async b128 store per iteration, no branches.
  unsigned long long ga =
      (unsigned long long)(uintptr_t)(rowp) + 16ull * threadIdx.x;
  for (int q = threadIdx.x; q < F4; q += 256) {
    async_store_b128_nt(ga, lds_off);
    ga += 256ull * 16ull;                           // block-wide stride
  }

  // Drain this wave's async stores, then the thread that issued the store
  // covering the target element (same wave => ordered) patches it to 0.
  wait_async_0();
  if (tgt >= 0 && (int)threadIdx.x == ((tgt >> 2) & 255)) {
    rowp[tgt] = 0.0f;
  }
}

extern "C" void kernel_launch(void* const* d_in, const int* in_sizes, int n_in,
                              void* d_out, int out_size, void* d_ws, size_t ws_size,
                              hipStream_t stream) {
  const long long* input_ids       = (const long long*)d_in[0]; // int64 (B,T)
  const int*       attention_mask  = (const int*)      d_in[1]; // int32 (B,T)
  const long long* start_positions = (const long long*)d_in[2]; // int64 (B,)
  float*           out             = (float*)d_out;             // f32 (B,T,V)

  const int B    = in_sizes[2];        // start_positions element count
  const int rows = in_sizes[1];        // B*T
  const int T    = rows / B;

  BoundaryAwareDummyCausalLM_row_fill<<<dim3(T, B), dim3(256), 0, stream>>>(
      input_ids, attention_mask, start_positions, out);

  (void)n_in; (void)out_size; (void)d_ws; (void)ws_size;
}